// Attention_75222057222310
// MI455X (gfx1250) — compile-verified
//
#include <hip/hip_runtime.h>
#include <hip/hip_bf16.h>
#include <math.h>

typedef __attribute__((ext_vector_type(2))) float v2f;
typedef __attribute__((ext_vector_type(4))) float f4;
typedef __attribute__((ext_vector_type(8))) float v8f;

#define BB 32
#define MM 2048
#define NN 1024   // 2*HID

// ---------------------------------------------------------------------------
// Kernel 1: dec_feature = h_c_hat @ Wd^T + bd   via V_WMMA_F32_16X16X4_F32
// Grid: (64 col-tiles, 2 row-tiles), 1 wave (32 threads) per tile.
// A (16x4 f32): lane L%16 = row M, VGPR0/1 = K = (L/16)*2 + {0,1}
// B (4x16 f32): B[k][n] = Wd[n][k]; lane L%16 = col n, VGPR0/1 = same K split
// C/D (16x16 f32): VGPR r -> row r (+8 for lanes 16..31), col = lane%16
// ---------------------------------------------------------------------------
__global__ __launch_bounds__(32) void k_decproj(const float* __restrict__ h,
                                                const float* __restrict__ Wd,
                                                const float* __restrict__ bd,
                                                float* __restrict__ dec) {
  const int colTile = blockIdx.x;      // 0..63
  const int rowTile = blockIdx.y;      // 0..1
  const int lane = threadIdx.x;
  const int lm = lane & 15;
  const int lh = lane >> 4;            // 0 or 1
  const float* __restrict__ arow = h  + (size_t)(rowTile * 16 + lm) * NN;
  const float* __restrict__ brow = Wd + (size_t)(colTile * 16 + lm) * NN;
  v8f acc = {};
  for (int k0 = 0; k0 < NN; k0 += 4) {
    const int ka = k0 + lh * 2;
    v2f a, b;
    a.x = arow[ka];     a.y = arow[ka + 1];
    b.x = brow[ka];     b.y = brow[ka + 1];
    acc = __builtin_amdgcn_wmma_f32_16x16x4_f32(
        /*neg_a=*/false, a, /*neg_b=*/false, b,
        /*c_mod=*/(short)0, acc, /*reuse_a=*/false, /*reuse_b=*/false);
  }
  const int col = colTile * 16 + lm;
  const float bias = bd[col];
#pragma unroll
  for (int r = 0; r < 8; ++r) {
    const int row = rowTile * 16 + r + lh * 8;
    dec[(size_t)row * NN + col] = acc[r] + bias;
  }
}

// ---------------------------------------------------------------------------
// Kernel 2: scores[b][m] = sum_n tanh(enc_feature + dec + cov*wc) * v
// 1 wave per (b,m) row; float4 nontemporal streaming of enc_feature (256 MB).
// tanh(x) = 1 - 2/(exp(2x)+1)  -> v_exp_f32 (handles +-inf saturation).
// ---------------------------------------------------------------------------
__global__ __launch_bounds__(256) void k_scores(const float* __restrict__ enc_feature,
                                                const float* __restrict__ dec,
                                                const float* __restrict__ coverage,
                                                const float* __restrict__ wc,
                                                const float* __restrict__ v,
                                                float* __restrict__ scores) {
  const int wave = threadIdx.x >> 5;
  const int lane = threadIdx.x & 31;
  const int row  = blockIdx.x * 8 + wave;   // 0 .. B*M-1
  const int b    = row >> 11;               // M = 2048
  const float cov = coverage[row];
  const float* __restrict__ ef = enc_feature + (size_t)row * NN;
  const float* __restrict__ db = dec + (size_t)b * NN;
  float s = 0.f;
#pragma unroll
  for (int k = 0; k < 8; ++k) {
    const int n = k * 128 + lane * 4;
    f4 e  = __builtin_nontemporal_load((const f4*)(ef + n));
    f4 d  = *(const f4*)(db + n);
    f4 w  = *(const f4*)(wc + n);
    f4 vv = *(const f4*)(v + n);
    f4 a  = e + d + cov * w;
#pragma unroll
    for (int i = 0; i < 4; ++i) {
      const float t = 1.f - 2.f / (__expf(2.f * a[i]) + 1.f);
      s += t * vv[i];
    }
  }
#pragma unroll
  for (int off = 16; off; off >>= 1) s += __shfl_xor(s, off, 32);
  if (lane == 0) scores[row] = s;
}

// ---------------------------------------------------------------------------
// Kernel 3: softmax over M, mask, renormalize; write attn and new_coverage.
// 1 block (256 threads = 8 waves) per batch; each thread owns 8 m's.
// ---------------------------------------------------------------------------
__global__ __launch_bounds__(256) void k_softmax(const float* __restrict__ scores,
                                                 const float* __restrict__ mask,
                                                 const float* __restrict__ coverage,
                                                 float* __restrict__ attn,
                                                 float* __restrict__ newcov) {
  __shared__ float sm[8];
  const int b = blockIdx.x;
  const int t = threadIdx.x;
  const int lane = t & 31, wid = t >> 5;
  const float* __restrict__ sc = scores + (size_t)b * MM;

  float vals[8];
  float mx = -INFINITY;
#pragma unroll
  for (int i = 0; i < 8; ++i) { vals[i] = sc[t + i * 256]; mx = fmaxf(mx, vals[i]); }
#pragma unroll
  for (int off = 16; off; off >>= 1) mx = fmaxf(mx, __shfl_xor(mx, off, 32));
  if (lane == 0) sm[wid] = mx;
  __syncthreads();
  float m0 = sm[0];
#pragma unroll
  for (int i = 1; i < 8; ++i) m0 = fmaxf(m0, sm[i]);
  __syncthreads();

  float es[8]; float lsum = 0.f;
#pragma unroll
  for (int i = 0; i < 8; ++i) { es[i] = __expf(vals[i] - m0); lsum += es[i]; }
#pragma unroll
  for (int off = 16; off; off >>= 1) lsum += __shfl_xor(lsum, off, 32);
  if (lane == 0) sm[wid] = lsum;
  __syncthreads();
  float tot = 0.f;
#pragma unroll
  for (int i = 0; i < 8; ++i) tot += sm[i];
  __syncthreads();

  const float inv = 1.f / tot;
  float ms[8]; float lsum2 = 0.f;
#pragma unroll
  for (int i = 0; i < 8; ++i) {
    const float p  = es[i] * inv;
    const float mk = mask[(size_t)b * MM + t + i * 256];
    ms[i] = p * (1.f - mk);
    lsum2 += ms[i];
  }
#pragma unroll
  for (int off = 16; off; off >>= 1) lsum2 += __shfl_xor(lsum2, off, 32);
  if (lane == 0) sm[wid] = lsum2;
  __syncthreads();
  float tot2 = 0.f;
#pragma unroll
  for (int i = 0; i < 8; ++i) tot2 += sm[i];
  const float inv2 = 1.f / tot2;
#pragma unroll
  for (int i = 0; i < 8; ++i) {
    const size_t idx = (size_t)b * MM + t + i * 256;
    const float a = ms[i] * inv2;
    attn[idx]   = a;
    newcov[idx] = coverage[idx] + a;
  }
}

// ---------------------------------------------------------------------------
// Kernel 4: partial context = sum over m-chunk of attn[b][m]*enc[b][m][:]
// Grid (8 m-chunks of 256, B); block 256 threads, float4 per thread (full N).
// ---------------------------------------------------------------------------
__global__ __launch_bounds__(256) void k_ctx_partial(const float* __restrict__ attn,
                                                     const float* __restrict__ enc,
                                                     float* __restrict__ partial) {
  const int b  = blockIdx.y;
  const int mc = blockIdx.x;            // 0..7, 256 rows each
  const int n4 = threadIdx.x;           // 256 float4 lanes = N
  const float* __restrict__ at = attn + (size_t)b * MM + mc * 256;
  const f4* __restrict__ e =
      (const f4*)(enc + ((size_t)b * MM + (size_t)mc * 256) * NN) + n4;
  f4 acc = {0.f, 0.f, 0.f, 0.f};
  for (int m = 0; m < 256; ++m) {
    const float a = at[m];
    const f4 ev = __builtin_nontemporal_load(e + (size_t)m * (NN / 4));
    acc += a * ev;
  }
  ((f4*)partial)[((size_t)(b * 8 + mc)) * (NN / 4) + n4] = acc;
}

// Kernel 5: fold 8 partials into context_vec [B, N]
__global__ __launch_bounds__(256) void k_ctx_reduce(const float* __restrict__ partial,
                                                    float* __restrict__ ctx) {
  const int i = blockIdx.x * 256 + threadIdx.x;   // 0..B*N-1
  const int b = i >> 10, n = i & 1023;
  float s = 0.f;
#pragma unroll
  for (int j = 0; j < 8; ++j) s += partial[((size_t)(b * 8 + j)) * NN + n];
  ctx[i] = s;
}

extern "C" void kernel_launch(void* const* d_in, const int* in_sizes, int n_in,
                              void* d_out, int out_size, void* d_ws, size_t ws_size,
                              hipStream_t stream) {
  const float* h_c_hat     = (const float*)d_in[0];  // [B, N]
  const float* enc_outputs = (const float*)d_in[1];  // [B, M, N]
  const float* enc_feature = (const float*)d_in[2];  // [B*M, N]
  const float* mask        = (const float*)d_in[3];  // [B, M]
  const float* coverage    = (const float*)d_in[4];  // [B, M]
  const float* Wd          = (const float*)d_in[5];  // [N, N]
  const float* bd          = (const float*)d_in[6];  // [N]
  const float* wc          = (const float*)d_in[7];  // [N]
  const float* v           = (const float*)d_in[8];  // [N]

  float* out    = (float*)d_out;
  float* ctx    = out;                 // [B, N]    = 32768
  float* attn   = out + BB * NN;       // [B, M]    = 65536
  float* newcov = attn + BB * MM;      // [B, M]    = 65536

  float* ws      = (float*)d_ws;
  float* dec     = ws;                 // B*N       = 32768 floats
  float* scores  = dec + BB * NN;      // B*M       = 65536 floats
  float* partial = scores + BB * MM;   // B*8*N     = 262144 floats (~1.4 MB total)

  k_decproj<<<dim3(NN / 16, BB / 16), 32, 0, stream>>>(h_c_hat, Wd, bd, dec);
  k_scores<<<(BB * MM) / 8, 256, 0, stream>>>(enc_feature, dec, coverage, wc, v, scores);
  k_softmax<<<BB, 256, 0, stream>>>(scores, mask, coverage, attn, newcov);
  k_ctx_partial<<<dim3(8, BB), 256, 0, stream>>>(attn, enc_outputs, partial);
  k_ctx_reduce<<<(BB * NN) / 256, 256, 0, stream>>>(partial, ctx);
}